// Matcher_73693048865555
// MI455X (gfx1250) — compile-verified
//
#include <hip/hip_runtime.h>
#include <hip/hip_bf16.h>
#include <math.h>

#define PI_F      3.141592653f
#define TWO_PI_F  (2.0f * PI_F)
#define IOU_THR_F 0.3f
// true overlap bound: sum of half-diagonals <= sqrt(4.5^2+2.0^2) ~ 4.93 m.
// 7 m cutoff is conservative (exact f32 distances from the WMMA path).
#define DIST2_CUT 49.0f

typedef __attribute__((ext_vector_type(2))) float v2f;
typedef __attribute__((ext_vector_type(8))) float v8f;

__device__ __forceinline__ float limit_period_f(float v) {
  return v - floorf(v / TWO_PI_F + 0.5f) * TWO_PI_F;
}

// ---------------- kernel 1: heading normalization ----------------
__global__ void prep_kernel(const float* __restrict__ in,
                            float* __restrict__ boxes, int n) {
  int i = blockIdx.x * blockDim.x + threadIdx.x;
  if (i < n) {
#pragma unroll
    for (int c = 0; c < 6; ++c) boxes[i * 7 + c] = in[i * 7 + c];
    boxes[i * 7 + 6] = limit_period_f(in[i * 7 + 6]);
  }
}

// ---------------- kernel 2: WMMA distance matrix ----------------
// D = A x B with K=4:
//   A row i  = [x_i, y_i, |c_i|^2, 1]
//   B col j  = [-2x_j, -2y_j, 1, |c_j|^2]
//   D[i][j]  = |c_i - c_j|^2   (exact f32)
// One wave per 16x128 strip: A fragment hoisted, 8 unrolled WMMAs.
// EXEC all-ones (blockDim.x == 32, no divergence before the WMMAs).
// 32-bit A 16x4 layout: lanes 0-15 hold K0/K1, lanes 16-31 hold K2/K3;
// 32-bit B 4x16 layout mirrors A with N across lanes.
__global__ void dist_wmma_kernel(const float* __restrict__ boxes,
                                 float* __restrict__ mat, int n) {
  const int lane = threadIdx.x;          // 0..31
  const int bi  = blockIdx.y;            // M tile
  const int bj0 = blockIdx.x * 8;        // first of 8 N tiles
  const int m   = lane & 15;
  const bool hi = lane >= 16;

  const int i = bi * 16 + m;
  const float xi = boxes[i * 7 + 0], yi = boxes[i * 7 + 1];
  const float sqi = xi * xi + yi * yi;
  v2f a;
  a.x = hi ? sqi  : xi;    // K=0 (lo) / K=2 (hi)
  a.y = hi ? 1.0f : yi;    // K=1 (lo) / K=3 (hi)

  const int rbase = bi * 16 + (hi ? 8 : 0);
#pragma unroll
  for (int t = 0; t < 8; ++t) {
    const int j = (bj0 + t) * 16 + m;
    const float xj = boxes[j * 7 + 0], yj = boxes[j * 7 + 1];
    const float sqj = xj * xj + yj * yj;
    v2f b;
    b.x = hi ? 1.0f : -2.0f * xj;  // K=0 (lo) / K=2 (hi)
    b.y = hi ? sqj  : -2.0f * yj;  // K=1 (lo) / K=3 (hi)

    v8f c = {};
    v8f d = __builtin_amdgcn_wmma_f32_16x16x4_f32(
        /*neg_a=*/false, a, /*neg_b=*/false, b,
        /*c_mod=*/(short)0, c, /*reuse_a=*/false, /*reuse_b=*/false);

    // C/D layout: VGPR r, lanes 0-15 -> (M=r, N=lane); lanes 16-31 -> M=r+8
    const int col = (bj0 + t) * 16 + m;
#pragma unroll
    for (int r = 0; r < 8; ++r) {
      mat[(size_t)(rbase + r) * n + col] = d[r];
    }
  }
}

// ---------------- kernel 3: full IoU for surviving pairs ----------------
__global__ void iou_kernel(const float* __restrict__ boxes,
                           float* __restrict__ mat, int n) {
  const long long idx = (long long)blockIdx.x * blockDim.x + threadIdx.x;
  const long long total = (long long)n * n;
  if (idx >= total) return;
  const int i = (int)(idx / n), j = (int)(idx % n);

  const float d2 = mat[idx];
  if (d2 > DIST2_CUT) { mat[idx] = 0.0f; return; }

  const float* ba = boxes + (size_t)i * 7;
  const float* bb = boxes + (size_t)j * 7;

  // BEV corners: signs * [dx,dy], rotated by rot_t = [[c,s],[-s,c]], + center
  const float sgx[4] = {0.5f, 0.5f, -0.5f, -0.5f};
  const float sgy[4] = {-0.5f, 0.5f, 0.5f, -0.5f};
  float pax[4], pay[4], pbx[4], pby[4];
  {
    const float cA = cosf(ba[6]), sA = sinf(ba[6]);
#pragma unroll
    for (int k = 0; k < 4; ++k) {
      const float lx = sgx[k] * ba[3], ly = sgy[k] * ba[4];
      pax[k] = lx * cA - ly * sA + ba[0];
      pay[k] = lx * sA + ly * cA + ba[1];
    }
    const float cB = cosf(bb[6]), sB = sinf(bb[6]);
#pragma unroll
    for (int k = 0; k < 4; ++k) {
      const float lx = sgx[k] * bb[3], ly = sgy[k] * bb[4];
      pbx[k] = lx * cB - ly * sB + bb[0];
      pby[k] = lx * sB + ly * cB + bb[1];
    }
  }

  // Sutherland-Hodgman: clip A's quad by B's 4 half-planes
  float px[8], py[8];
  int cnt = 4;
#pragma unroll
  for (int v = 0; v < 8; ++v) {
    px[v] = (v < 4) ? pax[v] : 0.0f;
    py[v] = (v < 4) ? pay[v] : 0.0f;
  }
#pragma unroll
  for (int k = 0; k < 4; ++k) {
    const float ax = pbx[k], ay = pby[k];
    const float ex = pbx[(k + 1) & 3] - ax;
    const float ey = pby[(k + 1) & 3] - ay;
    float nx2[8], ny2[8];
    int nc = 0;
    for (int v = 0; v < cnt; ++v) {
      const int w = (v + 1 < cnt) ? v + 1 : 0;
      const float d1 = ex * (py[v] - ay) - ey * (px[v] - ax);
      const float dB = ex * (py[w] - ay) - ey * (px[w] - ax);
      const bool in1 = d1 >= 0.0f, in2 = dB >= 0.0f;
      if (in1 && nc < 8) { nx2[nc] = px[v]; ny2[nc] = py[v]; ++nc; }
      if ((in1 != in2) && nc < 8) {
        const float den = d1 - dB;
        const float t = d1 / (fabsf(den) > 1e-8f ? den : 1.0f);
        nx2[nc] = px[v] + t * (px[w] - px[v]);
        ny2[nc] = py[v] + t * (py[w] - py[v]);
        ++nc;
      }
    }
#pragma unroll
    for (int v = 0; v < 8; ++v) {
      px[v] = (v < nc) ? nx2[v] : 0.0f;
      py[v] = (v < nc) ? ny2[v] : 0.0f;
    }
    cnt = nc;
  }

  float area2 = 0.0f;
#pragma unroll
  for (int v = 0; v < 8; ++v) {
    if (v < cnt) {
      const int w = (v + 1 < cnt) ? v + 1 : 0;
      area2 += px[v] * py[w] - px[w] * py[v];
    }
  }
  const float inter_bev = 0.5f * fabsf(area2);
  const float top = fminf(ba[2] + 0.5f * ba[5], bb[2] + 0.5f * bb[5]);
  const float bot = fmaxf(ba[2] - 0.5f * ba[5], bb[2] - 0.5f * bb[5]);
  const float inter = inter_bev * fmaxf(top - bot, 0.0f);
  const float va = ba[3] * ba[4] * ba[5];
  const float vb = bb[3] * bb[4] * bb[5];
  mat[idx] = inter / fmaxf(va + vb - inter, 1e-6f);
}

// ---------------- kernel 4: sequential greedy clustering ----------------
// Reference semantics: iteration `it` uses cur == it+1 (cur increments on
// every iteration that still has an unassigned box); once no box is
// unassigned, all remaining reference iterations are no-ops -> break early.
// Double-buffered sh_first: 2 barriers per iteration.
__global__ void cluster_kernel(const float* __restrict__ iou,
                               int* __restrict__ ci_out, int n) {
  __shared__ int ci_s[1024];
  __shared__ int sh_first[2];
  const int tid = threadIdx.x;
  ci_s[tid] = 0;
  if (tid < 2) sh_first[tid] = 0x7FFFFFFF;
  __syncthreads();
  for (int it = 0; it < n; ++it) {
    const int slot = it & 1;
    if (ci_s[tid] == 0) atomicMin(&sh_first[slot], tid);
    __syncthreads();
    const int first = sh_first[slot];          // uniform
    if (first == 0x7FFFFFFF) break;            // uniform: done forever
    if (tid == 0) sh_first[slot ^ 1] = 0x7FFFFFFF;  // reset for next iter
    if (iou[(size_t)first * n + tid] > IOU_THR_F) ci_s[tid] = it + 1;
    __syncthreads();
  }
  ci_out[tid] = ci_s[tid];
}

// ---------------- kernel 5: cluster fusion ----------------
__global__ void fusion_kernel(const float* __restrict__ boxes,
                              const float* __restrict__ scores,
                              const int* __restrict__ ci,
                              float* __restrict__ out, int n) {
  const int cid = blockIdx.x + 1;
  const int tid = threadIdx.x;
  const int T = blockDim.x;  // 256
  __shared__ float redf[256];
  __shared__ int redi[256];

  // ---- stage 1: sum of masked scores, any(mask), argmax (first-index ties)
  float sum_loc = 0.0f;
  float best = -__builtin_inff();
  int bidx = n;
  int anyc = 0;
  for (int j = tid; j < n; j += T) {
    const bool m2 = (ci[j] == cid);
    const float sc = scores[j];
    sum_loc += m2 ? sc : 0.0f;
    anyc |= m2 ? 1 : 0;
    const float v = m2 ? sc : -__builtin_inff();
    if (v > best || (v == best && j < bidx)) { best = v; bidx = j; }
  }
  redf[tid] = sum_loc; __syncthreads();
  for (int s2 = T >> 1; s2 > 0; s2 >>= 1) {
    if (tid < s2) redf[tid] += redf[tid + s2];
    __syncthreads();
  }
  const float sum_s = redf[0]; __syncthreads();

  redi[tid] = anyc; __syncthreads();
  for (int s2 = T >> 1; s2 > 0; s2 >>= 1) {
    if (tid < s2) redi[tid] |= redi[tid + s2];
    __syncthreads();
  }
  const int valid = redi[0]; __syncthreads();

  redf[tid] = best; redi[tid] = bidx; __syncthreads();
  for (int s2 = T >> 1; s2 > 0; s2 >>= 1) {
    if (tid < s2) {
      const float vo = redf[tid + s2]; const int io = redi[tid + s2];
      if (vo > redf[tid] || (vo == redf[tid] && io < redi[tid])) {
        redf[tid] = vo; redi[tid] = io;
      }
    }
    __syncthreads();
  }
  int amax = redi[0]; __syncthreads();
  if (amax >= n) amax = 0;  // all -inf -> jnp.argmax == 0
  const float d0 = boxes[(size_t)amax * 7 + 6];

  // ---- stage 2: direction-flip vote
  float sgt_loc = 0.0f, sngt_loc = 0.0f;
  for (int j = tid; j < n; j += T) {
    const bool m2 = (ci[j] == cid);
    const float s = m2 ? scores[j] : 0.0f;
    const float dir = boxes[(size_t)j * 7 + 6];
    float diff = fabsf(dir - d0);
    diff = (diff > PI_F) ? (TWO_PI_F - diff) : diff;
    if (diff > 0.5f * PI_F) sgt_loc += s; else sngt_loc += s;
  }
  redf[tid] = sgt_loc; __syncthreads();
  for (int s2 = T >> 1; s2 > 0; s2 >>= 1) {
    if (tid < s2) redf[tid] += redf[tid + s2];
    __syncthreads();
  }
  const float sgt = redf[0]; __syncthreads();
  redf[tid] = sngt_loc; __syncthreads();
  for (int s2 = T >> 1; s2 > 0; s2 >>= 1) {
    if (tid < s2) redf[tid] += redf[tid + s2];
    __syncthreads();
  }
  const float sngt = redf[0]; __syncthreads();
  const bool flip_gt = (sgt <= sngt);

  // ---- stage 3: weighted circular mean + weighted center/dims
  float acc[8];
#pragma unroll
  for (int q = 0; q < 8; ++q) acc[q] = 0.0f;
  const float inv = valid ? sum_s : 1.0f;
  for (int j = tid; j < n; j += T) {
    const bool m2 = (ci[j] == cid);
    const float s = m2 ? scores[j] : 0.0f;
    const float dir = boxes[(size_t)j * 7 + 6];
    float diff = fabsf(dir - d0);
    diff = (diff > PI_F) ? (TWO_PI_F - diff) : diff;
    const bool gt = diff > 0.5f * PI_F;
    const bool fm = flip_gt ? gt : !gt;
    float dd = fm ? dir + PI_F : dir;
    dd = limit_period_f(dd);
    const float sn = s / inv;
    acc[0] += sinf(dd) * sn;
    acc[1] += cosf(dd) * sn;
#pragma unroll
    for (int c = 0; c < 6; ++c) acc[2 + c] += boxes[(size_t)j * 7 + c] * sn;
  }
  float res[8];
  for (int q = 0; q < 8; ++q) {
    redf[tid] = acc[q]; __syncthreads();
    for (int s2 = T >> 1; s2 > 0; s2 >>= 1) {
      if (tid < s2) redf[tid] += redf[tid + s2];
      __syncthreads();
    }
    res[q] = redf[0]; __syncthreads();
  }

  if (tid == 0) {
    float* o = out + (size_t)(cid - 1) * 7;
    if (valid) {
#pragma unroll
      for (int c = 0; c < 6; ++c) o[c] = res[2 + c];
      o[6] = atan2f(res[0], res[1]);
    } else {
#pragma unroll
      for (int c = 0; c < 7; ++c) o[c] = 0.0f;
    }
  }
}

extern "C" void kernel_launch(void* const* d_in, const int* in_sizes, int n_in,
                              void* d_out, int out_size, void* d_ws, size_t ws_size,
                              hipStream_t stream) {
  const float* pred_boxes  = (const float*)d_in[0];
  const float* pred_scores = (const float*)d_in[1];
  float* out = (float*)d_out;
  const int n = in_sizes[0] / 7;  // 1024

  // workspace layout (256B-aligned sections):
  //   boxes_n : n*7 f32       (normalized boxes)
  //   ci      : n   i32       (cluster indices)
  //   mat     : n*n f32       (dist^2, overwritten in-place with IoU)
  char* ws = (char*)d_ws;
  float* boxes_n = (float*)ws;
  size_t off = (((size_t)n * 7 * sizeof(float)) + 255) & ~(size_t)255;
  int* ci = (int*)(ws + off);
  off += (((size_t)n * sizeof(int)) + 255) & ~(size_t)255;
  float* mat = (float*)(ws + off);

  prep_kernel<<<(n + 255) / 256, 256, 0, stream>>>(pred_boxes, boxes_n, n);

  dim3 gdist(n / 128, n / 16);  // one wave per 16x128 strip (8 WMMAs)
  dist_wmma_kernel<<<gdist, 32, 0, stream>>>(boxes_n, mat, n);

  const long long total = (long long)n * n;
  iou_kernel<<<(int)((total + 255) / 256), 256, 0, stream>>>(boxes_n, mat, n);

  cluster_kernel<<<1, n, 0, stream>>>(mat, ci, n);

  fusion_kernel<<<n, 256, 0, stream>>>(boxes_n, pred_scores, ci, out, n);
}